// First_Linear_Geometry_6871947673781
// MI455X (gfx1250) — compile-verified
//
#include <hip/hip_runtime.h>
#include <hip/hip_bf16.h>

typedef __attribute__((ext_vector_type(2))) float v2f;
typedef __attribute__((ext_vector_type(8))) float v8f;

#define KR   25088   // 512*7*7, K of the R-GEMM and weight row stride
#define NOUT 1000    // C_out
#define NB   256     // batch
#define CIN  512
#define HW   49

// ---------------------------------------------------------------------------
// Kernel 1: outputR[n,o] = sum_k inputR[n,k] * W[o,k] + bias[o]
// One wave per 16x32 output tile: two accumulators share one A-fragment,
// cutting L2 traffic to 1.5 loads per WMMA. Grid = 32 N-tiles x 16 M-tiles.
// A-frag lane layout (16x16x4 f32): M = lane&15, K pair = 2*(lane>>4) -> float2
// B-frag mirrors with N = lane&15. Both are contiguous 8B loads.
// ---------------------------------------------------------------------------
__global__ __launch_bounds__(32)
void gemmR_kernel(const float* __restrict__ A, const float* __restrict__ W,
                  const float* __restrict__ bias, float* __restrict__ out) {
  const int lane = threadIdx.x;
  const int half = lane >> 4;       // which K-pair this lane holds
  const int l16  = lane & 15;
  const int ntile = blockIdx.x;     // 0..31 (32-wide tiles)
  const int mtile = blockIdx.y;     // 0..15

  const int o0 = ntile * 32 + l16;
  const int o1 = o0 + 16;
  const int oc0 = (o0 < NOUT) ? o0 : 0;                 // branchless clamps (EXEC stays full)
  const int oc1 = (o1 < NOUT) ? o1 : 0;
  const float* arow  = A + (size_t)(mtile * 16 + l16) * KR + 2 * half;
  const float* brow0 = W + (size_t)oc0 * KR + 2 * half;
  const float* brow1 = W + (size_t)oc1 * KR + 2 * half;

  v8f acc0 = {}, acc1 = {};
  for (int k = 0; k < KR; k += 16) {
#pragma unroll
    for (int kk = 0; kk < 16; kk += 4) {
      v2f a  = *(const v2f*)(arow  + k + kk);
      v2f b0 = *(const v2f*)(brow0 + k + kk);
      v2f b1 = *(const v2f*)(brow1 + k + kk);
      acc0 = __builtin_amdgcn_wmma_f32_16x16x4_f32(false, a, false, b0,
                                                   (short)0, acc0, false, false);
      acc1 = __builtin_amdgcn_wmma_f32_16x16x4_f32(false, a, false, b1,
                                                   (short)0, acc1, false, false);
    }
  }

  const float bv0 = bias[oc0];
  const float bv1 = bias[oc1];
#pragma unroll
  for (int r = 0; r < 8; ++r) {                          // C/D: VGPR r -> M = r + 8*half
    const int row = mtile * 16 + r + 8 * half;
    if (o0 < NOUT) out[(size_t)row * NOUT + o0] = acc0[r] + bv0;
    if (o1 < NOUT) out[(size_t)row * NOUT + o1] = acc1[r] + bv1;
  }
}

// ---------------------------------------------------------------------------
// Kernel 2: per-row top-2 indices; also zero-init outputSPD (d_out is poisoned).
// ---------------------------------------------------------------------------
__global__ void top2_kernel(const float* __restrict__ outR, int* __restrict__ idx,
                            float* __restrict__ outSPD) {
  const int n = blockIdx.x * blockDim.x + threadIdx.x;
  if (n >= NB) return;
  const float* row = outR + (size_t)n * NOUT;
  float v1 = -3.402823466e38f, v2 = -3.402823466e38f;
  int i1 = 0, i2 = 0;
  for (int o = 0; o < NOUT; ++o) {
    const float v = row[o];
    if (v > v1)      { v2 = v1; i2 = i1; v1 = v; i1 = o; }
    else if (v > v2) { v2 = v;  i2 = o; }
  }
  idx[n * 2 + 0] = i1;
  idx[n * 2 + 1] = i2;
  outSPD[n * 4 + 0] = 0.f; outSPD[n * 4 + 1] = 0.f;
  outSPD[n * 4 + 2] = 0.f; outSPD[n * 4 + 3] = 0.f;
}

// ---------------------------------------------------------------------------
// Kernel 3: outputSPD[n,k,m] = sum_{c,hw} U_k[c,hw] * W[i_m][c,hw],
//           U_k = S[n] (512x512) @ W[i_k] (512x49 over d, hw padded to 64)
// One wave per (n, ctile): 8 accumulators (2 ksel x 4 hw-tiles) share each
// S-fragment, so the 268 MB inputSPD stream is read exactly once.
// hw-tile 3 clamps to column 48 for every lane (min(48+l16,48)==48); only the
// epilogue masks the invalid lanes.
// ---------------------------------------------------------------------------
__global__ __launch_bounds__(32)
void spd_kernel(const float* __restrict__ S, const float* __restrict__ W,
                const int* __restrict__ idx, float* __restrict__ outSPD) {
  const int lane = threadIdx.x;
  const int half = lane >> 4;
  const int l16  = lane & 15;
  const int ctile = blockIdx.x;        // 0..31
  const int n     = blockIdx.y;        // 0..255

  const int i0 = idx[n * 2 + 0];
  const int i1 = idx[n * 2 + 1];

  const int hwt[4] = { l16, 16 + l16, 32 + l16, 48 };    // tile 3: constant clamp
  const float* arow = S + (size_t)n * CIN * CIN
                        + (size_t)(ctile * 16 + l16) * CIN + 2 * half;
  // B lane-base pointers: element (d0, hw) sits at base + d*49 with
  // base = W + ik*KR + hw + half*98   (d0 = d + 2*half folded into base)
  const float* wk0 = W + (size_t)i0 * KR + (size_t)half * 2 * HW;
  const float* wk1 = W + (size_t)i1 * KR + (size_t)half * 2 * HW;

  v8f acc[2][4] = {};
#pragma unroll 2
  for (int d = 0; d < CIN; d += 4) {
    const v2f a = *(const v2f*)(arow + d);
    const size_t off = (size_t)d * HW;
#pragma unroll
    for (int t = 0; t < 4; ++t) {
      v2f b0, b1;
      b0.x = wk0[off + hwt[t]];        b0.y = wk0[off + HW + hwt[t]];
      b1.x = wk1[off + hwt[t]];        b1.y = wk1[off + HW + hwt[t]];
      acc[0][t] = __builtin_amdgcn_wmma_f32_16x16x4_f32(false, a, false, b0,
                                                        (short)0, acc[0][t], false, false);
      acc[1][t] = __builtin_amdgcn_wmma_f32_16x16x4_f32(false, a, false, b1,
                                                        (short)0, acc[1][t], false, false);
    }
  }

  // Fused Frobenius dots against both selected weight slices.
  const float* wm0 = W + (size_t)i0 * KR;
  const float* wm1 = W + (size_t)i1 * KR;
  float p00 = 0.f, p01 = 0.f, p10 = 0.f, p11 = 0.f;
#pragma unroll
  for (int t = 0; t < 4; ++t) {
    const float mask = (t < 3 || l16 == 0) ? 1.f : 0.f;  // tile 3 valid only at hw==48
#pragma unroll
    for (int r = 0; r < 8; ++r) {
      const int c = ctile * 16 + r + 8 * half;
      const float wA = mask * wm0[(size_t)c * HW + hwt[t]];
      const float wB = mask * wm1[(size_t)c * HW + hwt[t]];
      p00 += acc[0][t][r] * wA;  p01 += acc[0][t][r] * wB;
      p10 += acc[1][t][r] * wA;  p11 += acc[1][t][r] * wB;
    }
  }
  // wave32 reduction
  for (int off = 16; off > 0; off >>= 1) {
    p00 += __shfl_down(p00, off, 32);
    p01 += __shfl_down(p01, off, 32);
    p10 += __shfl_down(p10, off, 32);
    p11 += __shfl_down(p11, off, 32);
  }
  if (lane == 0) {
    atomicAdd(&outSPD[n * 4 + 0], p00);
    atomicAdd(&outSPD[n * 4 + 1], p01);
    atomicAdd(&outSPD[n * 4 + 2], p10);
    atomicAdd(&outSPD[n * 4 + 3], p11);
  }
}

// ---------------------------------------------------------------------------
extern "C" void kernel_launch(void* const* d_in, const int* in_sizes, int n_in,
                              void* d_out, int out_size, void* d_ws, size_t ws_size,
                              hipStream_t stream) {
  const float* inputR   = (const float*)d_in[0];   // (256, 512, 7, 7)
  const float* inputSPD = (const float*)d_in[1];   // (256, 512, 512)
  const float* weight   = (const float*)d_in[2];   // (1000, 512, 7, 7)
  const float* bias     = (const float*)d_in[3];   // (1000,)

  float* outR   = (float*)d_out;                   // (256, 1000)
  float* outSPD = outR + (size_t)NB * NOUT;        // (256, 2, 2)
  int*   idx    = (int*)d_ws;                      // (256, 2) top-2 indices

  dim3 gA((NOUT + 31) / 32, NB / 16);              // 32 x 16 tiles of 16x32
  gemmR_kernel<<<gA, 32, 0, stream>>>(inputR, weight, bias, outR);

  top2_kernel<<<dim3(NB / 64), 64, 0, stream>>>(outR, idx, outSPD);

  dim3 gC(CIN / 16, NB);                           // ctile x n
  spd_kernel<<<gC, 32, 0, stream>>>(inputSPD, weight, idx, outSPD);
}